// FractionalSTFT_42253888258226
// MI455X (gfx1250) — compile-verified
//
#include <hip/hip_runtime.h>
#include <hip/hip_bf16.h>

// FractionalSTFT -> ISTFT, restructured as:
//   frames = window * framed(reflect_pad(x))                (elementwise)
//   M[n,m] = sum_g P[n,g]*Wre[g,m] + Q[n,g]*Wim[g,m]        (folded sym, f32 WMMA GEMM)
//   y      = frames . M^T                                   (f32 WMMA GEMM)
//   out    = overlap_add(y) / clip(win_norm, 1e-8)          (fused elementwise)
// 206 GFLOP direct -> ~39 GFLOP folded. Compute-bound on fp32 WMMA; weights
// (~201MB) stream once (~9us @ 23.3TB/s) and mostly live in the 192MB L2.
// Double-buffered LDS tiles fed by GLOBAL_LOAD_ASYNC_TO_LDS (ASYNCcnt-tracked)
// so tile prefetch overlaps the v_wmma stream.

typedef __attribute__((ext_vector_type(2))) float v2f;
typedef __attribute__((ext_vector_type(8))) float v8f;
typedef __attribute__((ext_vector_type(4))) int   i32x4;

#define AS1 __attribute__((address_space(1)))
#define AS3 __attribute__((address_space(3)))

#if __has_builtin(__builtin_amdgcn_global_load_async_to_lds_b128) && \
    __has_builtin(__builtin_amdgcn_global_load_async_to_lds_b32)
#define HAVE_ASYNC 1
#else
#define HAVE_ASYNC 0
#endif

namespace cfg {
constexpr int NFFT  = 1024;
constexpr int HOP   = 256;
constexpr int LEN   = 131072;
constexpr int PAD   = 512;
constexpr int T     = 513;           // frames per (b,c)
constexpr int BATCH = 4;             // b*c
constexpr int MROWS = BATCH * T;     // 2052
constexpr int MPAD  = 2064;          // padded to multiple of 16
constexpr int F1    = 8193;          // encoder rows
constexpr int F2    = 16384;         // extended spectrum / decoder cols
constexpr int KP    = 8208;          // F1 padded to multiple of BK
constexpr int BM = 128, BN = 128, BK = 16;
constexpr int BKP = BK + 4;          // LDS row stride: 16B-aligned rows, 4*(5q%16)
                                     // is a permutation -> conflict-free frags
}

__device__ __forceinline__ v8f wmma4(v2f a, v2f b, v8f c) {
  // V_WMMA_F32_16X16X4_F32 : D = A(16x4) * B(4x16) + C(16x16), fp32
  return __builtin_amdgcn_wmma_f32_16x16x4_f32(
      /*neg_a=*/false, a, /*neg_b=*/false, b,
      /*c_mod=*/(short)0, c, /*reuse_a=*/false, /*reuse_b=*/false);
}

__device__ __forceinline__ void async_b128(const float* g, float* l) {
#if HAVE_ASYNC
  float* gm = const_cast<float*>(g);
  __builtin_amdgcn_global_load_async_to_lds_b128((AS1 i32x4*)gm, (AS3 i32x4*)l,
                                                 0, 0);
#else
  (void)g; (void)l;
#endif
}

__device__ __forceinline__ void async_b32(const float* g, float* l) {
#if HAVE_ASYNC
  float* gm = const_cast<float*>(g);
  __builtin_amdgcn_global_load_async_to_lds_b32((AS1 int*)gm, (AS3 int*)l,
                                                0, 0);
#else
  (void)g; (void)l;
#endif
}

__device__ __forceinline__ void wait_async() {
#if HAVE_ASYNC
#if __has_builtin(__builtin_amdgcn_s_wait_asynccnt)
  __builtin_amdgcn_s_wait_asynccnt(0);
#else
  asm volatile("s_wait_asynccnt 0x0" ::: "memory");
#endif
#endif
}

// ---------------- frame + window (reflect pad) ----------------
__global__ __launch_bounds__(256) void k_frames(const float* __restrict__ x,
                                                const float* __restrict__ win,
                                                float* __restrict__ frames) {
  int g = blockIdx.x * 256 + threadIdx.x;
  if (g >= cfg::MPAD * cfg::NFFT) return;
  int r = g >> 10, n = g & 1023;
  float v = 0.f;
  if (r < cfg::MROWS) {
    int bc = r / cfg::T;
    int t  = r - bc * cfg::T;
    int j  = t * cfg::HOP + n - cfg::PAD;
    if (j < 0) j = -j;
    if (j >= cfg::LEN) j = 2 * cfg::LEN - 2 - j;
    v = x[bc * cfg::LEN + j] * win[n];
  }
  frames[g] = v;
}

// ---------------- M = P*Wre + Q*Wim (P,Q folded on the fly) ----------------
__global__ __launch_bounds__(256) void k_gemm_mmat(const float* __restrict__ Wdr,
                                                   const float* __restrict__ Wdi,
                                                   const float* __restrict__ Wre,
                                                   const float* __restrict__ Wim,
                                                   float* __restrict__ Mmat) {
  __shared__ __align__(16) float AsP[2][cfg::BM][cfg::BKP];
  __shared__ __align__(16) float AsQ[2][cfg::BM][cfg::BKP];
  __shared__ __align__(16) float BsR[2][cfg::BN][cfg::BKP];
  __shared__ __align__(16) float BsI[2][cfg::BN][cfg::BKP];

  const int tid  = threadIdx.x;
  const int lane = tid & 31;
  const int wave = tid >> 5;
  const int half = lane >> 4;
  const int lq   = lane & 15;
  const int wm   = (wave >> 1) * 32;   // 0,32,64,96
  const int wn   = (wave & 1) * 64;    // 0,64
  const int m0   = blockIdx.y * cfg::BM;
  const int n0   = blockIdx.x * cfg::BN;

  v8f acc[2][4] = {};

  auto stage = [&](int b, int k0) {
    // A tiles: P/Q built on the fly (symmetry fold) -> VALU + ds_store path.
    #pragma unroll
    for (int e = tid; e < cfg::BM * cfg::BK; e += 256) {
      int i = e >> 4, k = e & 15;
      int n = m0 + i, g = k0 + k;
      float p = 0.f, q = 0.f;
      if (g < cfg::F1) {
        p =  Wdr[(size_t)n * cfg::F2 + g];
        q = -Wdi[(size_t)n * cfg::F2 + g];
      }
      if (g >= 1 && g < cfg::F1 - 1) {
        p += Wdr[(size_t)n * cfg::F2 + (cfg::F2 - g)];
        q += Wdi[(size_t)n * cfg::F2 + (cfg::F2 - g)];
      }
      AsP[b][i][k] = p;
      AsQ[b][i][k] = q;
    }
    // B tiles: transposed copy of encoder rows; per-lane contiguous global
    // reads, strided LDS writes -> async b32 copies (ASYNCcnt).
    #pragma unroll
    for (int e = tid; e < cfg::BN * cfg::BK; e += 256) {
      int j = e & 127, k = e >> 7;
      int g = k0 + k;
      if (g > cfg::F1 - 1) g = cfg::F1 - 1;  // clamp: A is exactly 0 there
      size_t go = (size_t)g * cfg::NFFT + n0 + j;
#if HAVE_ASYNC
      async_b32(&Wre[go], &BsR[b][j][k]);
      async_b32(&Wim[go], &BsI[b][j][k]);
#else
      BsR[b][j][k] = Wre[go];
      BsI[b][j][k] = Wim[go];
#endif
    }
  };

  auto compute = [&](int b) {
    #pragma unroll
    for (int ks = 0; ks < cfg::BK / 4; ++ks) {
      const int kb = ks * 4 + 2 * half;
      v2f aP[2], aQ[2], bR[4], bI[4];
      #pragma unroll
      for (int t = 0; t < 2; ++t) {
        int row = wm + t * 16 + lq;
        aP[t][0] = AsP[b][row][kb]; aP[t][1] = AsP[b][row][kb + 1];
        aQ[t][0] = AsQ[b][row][kb]; aQ[t][1] = AsQ[b][row][kb + 1];
      }
      #pragma unroll
      for (int j = 0; j < 4; ++j) {
        int col = wn + j * 16 + lq;
        bR[j][0] = BsR[b][col][kb]; bR[j][1] = BsR[b][col][kb + 1];
        bI[j][0] = BsI[b][col][kb]; bI[j][1] = BsI[b][col][kb + 1];
      }
      #pragma unroll
      for (int t = 0; t < 2; ++t)
        #pragma unroll
        for (int j = 0; j < 4; ++j) {
          acc[t][j] = wmma4(aP[t], bR[j], acc[t][j]);
          acc[t][j] = wmma4(aQ[t], bI[j], acc[t][j]);
        }
    }
  };

  stage(0, 0);
  wait_async();
  __syncthreads();
  int p = 0;
  for (int k0 = 0; k0 < cfg::KP; k0 += cfg::BK, p ^= 1) {
    if (k0 + cfg::BK < cfg::KP) stage(p ^ 1, k0 + cfg::BK);  // prefetch next
    compute(p);
    wait_async();
    __syncthreads();
  }

  #pragma unroll
  for (int t = 0; t < 2; ++t)
    #pragma unroll
    for (int j = 0; j < 4; ++j) {
      int row0 = m0 + wm + t * 16 + 8 * half;
      int col  = n0 + wn + j * 16 + lq;
      #pragma unroll
      for (int v = 0; v < 8; ++v)
        Mmat[(size_t)(row0 + v) * cfg::NFFT + col] = acc[t][j][v];
    }
}

// ---------------- y = A(MPAD x 1024) . W(1024 x 1024)^T ----------------
__global__ __launch_bounds__(256) void k_gemm_awt(const float* __restrict__ A,
                                                  const float* __restrict__ W,
                                                  float* __restrict__ C) {
  __shared__ __align__(16) float As[2][cfg::BM][cfg::BKP];
  __shared__ __align__(16) float Bs[2][cfg::BN][cfg::BKP];

  const int tid  = threadIdx.x;
  const int lane = tid & 31;
  const int wave = tid >> 5;
  const int half = lane >> 4;
  const int lq   = lane & 15;
  const int wm   = (wave >> 1) * 32;
  const int wn   = (wave & 1) * 64;
  const int m0   = blockIdx.y * cfg::BM;
  const int n0   = blockIdx.x * cfg::BN;
  const int K    = cfg::NFFT;

  v8f acc[2][4] = {};

  auto stage = [&](int b, int k0) {
    // Both tiles are contiguous 16B row chunks -> async b128 copies.
    #pragma unroll
    for (int e = tid; e < cfg::BM * cfg::BK / 4; e += 256) {
      int i = e >> 2, k4 = (e & 3) << 2;
      int row = m0 + i;
      if (row > cfg::MPAD - 1) row = cfg::MPAD - 1;  // clamp: rows unstored
#if HAVE_ASYNC
      async_b128(&A[(size_t)row * K + k0 + k4], &As[b][i][k4]);
#else
      const float4 v = *(const float4*)&A[(size_t)row * K + k0 + k4];
      As[b][i][k4 + 0] = v.x; As[b][i][k4 + 1] = v.y;
      As[b][i][k4 + 2] = v.z; As[b][i][k4 + 3] = v.w;
#endif
    }
    #pragma unroll
    for (int e = tid; e < cfg::BN * cfg::BK / 4; e += 256) {
      int j = e >> 2, k4 = (e & 3) << 2;
#if HAVE_ASYNC
      async_b128(&W[(size_t)(n0 + j) * K + k0 + k4], &Bs[b][j][k4]);
#else
      const float4 v = *(const float4*)&W[(size_t)(n0 + j) * K + k0 + k4];
      Bs[b][j][k4 + 0] = v.x; Bs[b][j][k4 + 1] = v.y;
      Bs[b][j][k4 + 2] = v.z; Bs[b][j][k4 + 3] = v.w;
#endif
    }
  };

  auto compute = [&](int b) {
    #pragma unroll
    for (int ks = 0; ks < cfg::BK / 4; ++ks) {
      const int kb = ks * 4 + 2 * half;
      v2f a[2], bb[4];
      #pragma unroll
      for (int t = 0; t < 2; ++t) {
        int row = wm + t * 16 + lq;
        a[t][0] = As[b][row][kb]; a[t][1] = As[b][row][kb + 1];
      }
      #pragma unroll
      for (int j = 0; j < 4; ++j) {
        int col = wn + j * 16 + lq;
        bb[j][0] = Bs[b][col][kb]; bb[j][1] = Bs[b][col][kb + 1];
      }
      #pragma unroll
      for (int t = 0; t < 2; ++t)
        #pragma unroll
        for (int j = 0; j < 4; ++j)
          acc[t][j] = wmma4(a[t], bb[j], acc[t][j]);
    }
  };

  stage(0, 0);
  wait_async();
  __syncthreads();
  int p = 0;
  for (int k0 = 0; k0 < K; k0 += cfg::BK, p ^= 1) {
    if (k0 + cfg::BK < K) stage(p ^ 1, k0 + cfg::BK);  // prefetch next
    compute(p);
    wait_async();
    __syncthreads();
  }

  #pragma unroll
  for (int t = 0; t < 2; ++t)
    #pragma unroll
    for (int j = 0; j < 4; ++j) {
      int row0 = m0 + wm + t * 16 + 8 * half;
      int col  = n0 + wn + j * 16 + lq;
      #pragma unroll
      for (int v = 0; v < 8; ++v) {
        int row = row0 + v;
        if (row < cfg::MPAD) C[(size_t)row * cfg::NFFT + col] = acc[t][j][v];
      }
    }
}

// ---------------- fused overlap-add + window-norm ----------------
__global__ __launch_bounds__(256) void k_ola(const float* __restrict__ y,
                                             const float* __restrict__ win,
                                             float* __restrict__ out) {
  int g = blockIdx.x * 256 + threadIdx.x;
  if (g >= cfg::BATCH * cfg::LEN) return;
  int bc = g >> 17;
  int l  = (g & (cfg::LEN - 1)) + cfg::PAD;
  int t1 = l >> 8;
  if (t1 > cfg::T - 1) t1 = cfg::T - 1;
  int t0 = (l - (cfg::NFFT - 1) + (cfg::HOP - 1)) >> 8;   // ceil((l-1023)/256)
  if (t0 < 0) t0 = 0;
  float a = 0.f, wsum = 0.f;
  for (int t = t0; t <= t1; ++t) {
    int n = l - (t << 8);
    a    += y[((size_t)(bc * cfg::T + t)) * cfg::NFFT + n];
    wsum += win[n];
  }
  out[g] = a / fmaxf(wsum, 1e-8f);
}

extern "C" void kernel_launch(void* const* d_in, const int* in_sizes, int n_in,
                              void* d_out, int out_size, void* d_ws, size_t ws_size,
                              hipStream_t stream) {
  (void)in_sizes; (void)n_in; (void)out_size; (void)ws_size;
  const float* x   = (const float*)d_in[0];
  const float* win = (const float*)d_in[1];
  const float* Wre = (const float*)d_in[2];  // (8193, 1024)
  const float* Wim = (const float*)d_in[3];  // (8193, 1024)
  const float* Wdr = (const float*)d_in[4];  // (1024, 16384)
  const float* Wdi = (const float*)d_in[5];  // (1024, 16384)
  float* out = (float*)d_out;

  float* ws     = (float*)d_ws;
  float* frames = ws;                                        // MPAD x 1024
  float* mmat   = frames + (size_t)cfg::MPAD * cfg::NFFT;    // 1024 x 1024
  float* ybuf   = mmat   + (size_t)cfg::NFFT * cfg::NFFT;    // MPAD x 1024

  k_frames<<<(cfg::MPAD * cfg::NFFT + 255) / 256, 256, 0, stream>>>(x, win, frames);
  k_gemm_mmat<<<dim3(cfg::NFFT / cfg::BN, cfg::NFFT / cfg::BM), 256, 0, stream>>>(
      Wdr, Wdi, Wre, Wim, mmat);
  k_gemm_awt<<<dim3(cfg::NFFT / cfg::BN, (cfg::MPAD + cfg::BM - 1) / cfg::BM), 256, 0,
               stream>>>(frames, mmat, ybuf);
  k_ola<<<(cfg::BATCH * cfg::LEN + 255) / 256, 256, 0, stream>>>(ybuf, win, out);
}